// PosNegSampler_31988916420777
// MI455X (gfx1250) — compile-verified
//
#include <hip/hip_runtime.h>
#include <hip/hip_bf16.h>
#include <stddef.h>

// ---------------- problem constants ----------------
constexpr int B_    = 1024;   // batch rows
constexpr int C_    = 768;    // feature dim
constexpr int KQ    = 7500;   // queue length
constexpr int TOPK  = 25;
constexpr int STRIDE = 7552;          // padded row stride for cos buffer (64-float aligned)
constexpr unsigned NSORT = 8192;      // pow2 >= KQ
constexpr int NT_N = (KQ + 15) / 16;  // 469 N-tiles
constexpr int NT_M = B_ / 16;         // 64  M-tiles

typedef float v2f __attribute__((ext_vector_type(2)));
typedef float v8f __attribute__((ext_vector_type(8)));

// ---------------- K1: label histogram -> pos_min / neg_min / neg_topk ----------------
__global__ void __launch_bounds__(1024)
label_stats_kernel(const int* __restrict__ label_q,
                   const int* __restrict__ label_queue,
                   int* __restrict__ scal) {
  __shared__ int cnt[32];
  __shared__ int gmin, gmax;
  const int tid = threadIdx.x;
  if (tid < 32) cnt[tid] = 0;
  if (tid == 0) { gmin = 0x7fffffff; gmax = 0; }
  __syncthreads();
  for (int k = tid; k < KQ; k += 1024)
    atomicAdd(&cnt[label_queue[k] & 31], 1);
  __syncthreads();
  // exactly B_ == 1024 threads: tid == batch row
  int pc = cnt[label_q[tid] & 31];
  atomicMin(&gmin, pc);
  atomicMax(&gmax, pc);
  __syncthreads();
  if (tid == 0) {
    int pos_min  = gmin;
    int neg_min  = KQ - gmax;
    int neg_topk = pos_min < TOPK ? pos_min : TOPK;
    scal[0] = pos_min;
    scal[1] = neg_min;
    scal[2] = neg_topk;
  }
}

// ---------------- K2: cos_sim = liner_q @ fq^T via V_WMMA_F32_16X16X4_F32 ----------------
// One wave per 16x16 output tile. A is [B,C] row-major, fq is [KQ,C] row-major,
// so B-matrix columns are fq rows -> both A and B fragments are contiguous float2 loads.
__global__ void __launch_bounds__(256)
gemm_wmma_f32_kernel(const float* __restrict__ A,
                     const float* __restrict__ FQ,
                     float* __restrict__ cosbuf) {
  const int wave   = (int)((blockIdx.x * blockDim.x + threadIdx.x) >> 5);
  const int lane   = threadIdx.x & 31;
  const int laneLo = lane & 15;
  const int laneHi = lane >> 4;

  const int tm  = wave / NT_N;
  const int tn  = wave % NT_N;
  const int row = tm * 16 + laneLo;         // A row for this lane (M = laneLo)
  const int col = tn * 16 + laneLo;         // output col / fq row (N = laneLo)
  const int bcol = col < KQ ? col : KQ - 1; // clamp OOB fq loads; store is guarded

  // 32-bit A 16x4: lanes0-15 hold K={0,1} in VGPR{0,1}; lanes16-31 hold K={2,3}.
  // B 4x16 mirrors: low lanes rows K={0,1}, high lanes rows K={2,3}. Both -> float2.
  const float* a_ptr = A  + (size_t)row  * C_ + 2 * laneHi;
  const float* b_ptr = FQ + (size_t)bcol * C_ + 2 * laneHi;

  v8f acc = {};
#pragma unroll 4
  for (int k0 = 0; k0 < C_; k0 += 4) {
    v2f av = *(const v2f*)(a_ptr + k0);
    v2f bv = *(const v2f*)(b_ptr + k0);
    acc = __builtin_amdgcn_wmma_f32_16x16x4_f32(
        /*neg_a=*/false, av, /*neg_b=*/false, bv,
        /*c_mod=*/(short)0, acc, /*reuse_a=*/false, /*reuse_b=*/false);
  }

  if (col < KQ) {
#pragma unroll
    for (int v = 0; v < 8; ++v) {
      int m = tm * 16 + laneHi * 8 + v;  // C/D layout: VGPR v -> M=v (lo lanes) / v+8 (hi)
      cosbuf[(size_t)m * STRIDE + col] = acc[v];
    }
  }
}

// ---------------- K3: per-row sort + compaction + emission ----------------
// Dynamic LDS carve (bytes):
constexpr unsigned OFF_KEYS = 0;                       // 8192 * 8  = 65536
constexpr unsigned OFF_NEG  = OFF_KEYS + NSORT * 8;    // 8192 * 4  = 32768
constexpr unsigned OFF_SC   = OFF_NEG  + NSORT * 4;    // 1024 * 4  =  4096
constexpr unsigned OFF_SSUM = OFF_SC   + 1024 * 4;     //   256
constexpr unsigned OFF_POS  = OFF_SSUM + 256;          //   256
constexpr unsigned OFF_LBL  = OFF_POS  + 256;          //  7552 (byte labels)
constexpr unsigned SMEM_BYTES = OFF_LBL + 7552;        // = 110464 < 320KB/WGP

__device__ __forceinline__ unsigned ord_encode(float f) {
  unsigned u = __float_as_uint(f);
  return (u & 0x80000000u) ? ~u : (u | 0x80000000u);
}
__device__ __forceinline__ float ord_decode(unsigned o) {
  unsigned u = (o & 0x80000000u) ? (o & 0x7fffffffu) : ~o;
  return __uint_as_float(u);
}

__global__ void __launch_bounds__(1024)
sort_emit_kernel(const float* __restrict__ cosbuf,
                 const int* __restrict__ label_q,
                 const int* __restrict__ label_queue,
                 const int* __restrict__ scal,
                 float* __restrict__ out) {
  extern __shared__ unsigned char smem[];
  unsigned long long* keys = (unsigned long long*)(smem + OFF_KEYS);
  float*  negbuf = (float*)(smem + OFF_NEG);
  int*    sc     = (int*)  (smem + OFF_SC);
  int*    ssum   = (int*)  (smem + OFF_SSUM);
  float*  posbuf = (float*)(smem + OFF_POS);
  unsigned char* lbl = (unsigned char*)(smem + OFF_LBL);

  const int tid = threadIdx.x;
  const int b   = blockIdx.x;
  const int neg_min  = scal[1];
  const int neg_topk = scal[2];
  const int myLabel  = label_q[b];

  for (int k = tid; k < KQ; k += 1024) lbl[k] = (unsigned char)label_queue[k];

  // encode row into order-preserving 64-bit keys (value<<32 | idx); pads sort last
  const float* rowp = cosbuf + (size_t)b * STRIDE;
  for (unsigned i = tid; i < NSORT; i += 1024) {
    if (i < (unsigned)KQ) {
      keys[i] = ((unsigned long long)ord_encode(rowp[i]) << 32) | i;
    } else {
      keys[i] = (unsigned long long)(NSORT - 1);  // ord=0 (minimal), idx>=KQ (excluded)
    }
  }
  __syncthreads();

  // bitonic sort, descending, 8192 elements, 4 disjoint pairs per thread per substep
  for (unsigned kk = 2; kk <= NSORT; kk <<= 1) {
    for (unsigned j = kk >> 1; j > 0; j >>= 1) {
      for (unsigned p = tid; p < NSORT / 2; p += 1024) {
        unsigned i = 2u * p - (p & (j - 1u));   // bit j clear
        unsigned l = i + j;
        unsigned long long x = keys[i], y = keys[l];
        bool up = ((i & kk) == 0);
        bool sw = up ? (x < y) : (x > y);
        if (sw) { keys[i] = y; keys[l] = x; }
      }
      __syncthreads();
    }
  }

  // decode my 8 contiguous sorted entries; flag negatives; value / T == value * 2
  int   flags[8];
  float vals[8];
  int cnt = 0;
#pragma unroll
  for (int e = 0; e < 8; ++e) {
    unsigned i = (unsigned)tid * 8u + (unsigned)e;
    unsigned long long kv = keys[i];
    unsigned ord = (unsigned)(kv >> 32);
    unsigned idx = (unsigned)kv;
    vals[e]  = ord_decode(ord) * 2.0f;
    flags[e] = (idx < (unsigned)KQ) && ((int)lbl[idx] != myLabel);
    cnt += flags[e];
  }
  if (tid < 32) posbuf[tid] = ord_decode((unsigned)(keys[tid] >> 32)) * 2.0f;

  // block-exclusive-scan of per-thread neg counts (32x32 two-level)
  sc[tid] = cnt;
  __syncthreads();
  if (tid < 32) {
    int s = 0;
    for (int q = 0; q < 32; ++q) { int t = sc[tid * 32 + q]; sc[tid * 32 + q] = s; s += t; }
    ssum[tid] = s;
  }
  __syncthreads();
  if (tid == 0) {
    int s = 0;
    for (int q = 0; q < 32; ++q) { int t = ssum[q]; ssum[q] = s; s += t; }
  }
  __syncthreads();
  int pos = sc[tid] + ssum[tid >> 5];
#pragma unroll
  for (int e = 0; e < 8; ++e) {
    if (flags[e]) { if (pos < neg_min) negbuf[pos] = vals[e]; ++pos; }
  }
  __syncthreads();

  // emit: row r = b*neg_topk + t : [pos_t, neg_0..neg_{neg_min-1}] (already /T)
  const size_t Lrow = 1 + (size_t)neg_min;
  for (int t = 0; t < neg_topk; ++t) {
    size_t base = ((size_t)b * (size_t)neg_topk + (size_t)t) * Lrow;
    if (tid == 0) out[base] = posbuf[t];
    for (int j = tid; j < neg_min; j += 1024) out[base + 1 + j] = negbuf[j];
  }
}

// ---------------- launch ----------------
extern "C" void kernel_launch(void* const* d_in, const int* in_sizes, int n_in,
                              void* d_out, int out_size, void* d_ws, size_t ws_size,
                              hipStream_t stream) {
  (void)in_sizes; (void)n_in; (void)out_size; (void)ws_size;
  const float* liner_q     = (const float*)d_in[0];
  const int*   label_q     = (const int*)  d_in[1];
  const float* fq          = (const float*)d_in[2];
  const int*   label_queue = (const int*)  d_in[3];
  float* out = (float*)d_out;

  int*   scal   = (int*)d_ws;
  float* cosbuf = (float*)((char*)d_ws + 256);

  label_stats_kernel<<<1, 1024, 0, stream>>>(label_q, label_queue, scal);

  const int total_tiles = NT_M * NT_N;            // 30016 waves
  gemm_wmma_f32_kernel<<<total_tiles / 8, 256, 0, stream>>>(liner_q, fq, cosbuf);

  sort_emit_kernel<<<B_, 1024, SMEM_BYTES, stream>>>(cosbuf, label_q, label_queue, scal, out);
}